// BilateralFiltering_32736240730432
// MI455X (gfx1250) — compile-verified
//
#include <hip/hip_runtime.h>
#include <math.h>

// Problem dims (fixed by reference): N=4 images, K=3 channels, 96x96 pixels.
#define NIMG 4
#define KCH 3
#define HH 96
#define WW 96
#define PP (HH * WW)            // 9216 pixels
#define TILE_I 64               // output pixels per workgroup (4 waves x 16)
#define NTILES (PP / TILE_I)    // 144
#define THREADS 128

typedef __attribute__((ext_vector_type(2))) float v2f;
typedef __attribute__((ext_vector_type(8))) float v8f;

__device__ __forceinline__ void atomic_max_float(float* addr, float val) {
    // Order-preserving float max for mixed signs via int/uint atomics.
    if (val >= 0.0f) atomicMax((int*)addr, __float_as_int(val));
    else             atomicMin((unsigned int*)addr, __float_as_uint(val));
}

__global__ void bf_init_max(float* mx) {
    if (threadIdx.x == 0 && blockIdx.x == 0) *mx = -INFINITY;
}

__global__ __launch_bounds__(THREADS) void bf_main(
    const float* __restrict__ raw,   // [N,3,96,96]
    const float* __restrict__ nrm,   // [N,3,96,96]
    float* __restrict__ out,         // [N,3,96,96] (unnormalized here)
    float* __restrict__ mx)          // global max scratch
{
    extern __shared__ float smem[];  // 7 * PP floats (252 KB)
    float* s_sq = smem;
    float* s_c0 = smem + PP;
    float* s_c1 = smem + 2 * PP;
    float* s_c2 = smem + 3 * PP;
    float* s_v0 = smem + 4 * PP;
    float* s_v1 = smem + 5 * PP;
    float* s_v2 = smem + 6 * PP;

    const int n    = blockIdx.x / NTILES;
    const int tile = blockIdx.x % NTILES;
    const int tid  = threadIdx.x;

    const float inv_rgb = 1.0f / 15.0f;   // 1/SIGMA_RGB
    const float inv_xy  = 1.0f / 100.0f;  // 1/SIGMA_XY

    const float* rawn = raw + (size_t)n * KCH * PP;
    const float* nrmn = nrm + (size_t)n * KCH * PP;

    // ---- Stage features, squared norms and value channels into LDS ----
    for (int p = tid; p < PP; p += THREADS) {          // 9216/128 = 72, no tail
        float c0 = rawn[p]          * inv_rgb;
        float c1 = rawn[PP + p]     * inv_rgb;
        float c2 = rawn[2 * PP + p] * inv_rgb;
        float x  = (float)(p % WW) * inv_xy;
        float y  = (float)(p / WW) * inv_xy;
        s_c0[p] = c0; s_c1[p] = c1; s_c2[p] = c2;
        s_sq[p] = x * x + y * y + c0 * c0 + c1 * c1 + c2 * c2;
        s_v0[p] = nrmn[p];
        s_v1[p] = nrmn[PP + p];
        s_v2[p] = nrmn[2 * PP + p];
    }
    __syncthreads();

    const int lane = tid & 31;
    const int l16  = lane & 15;
    const int hl   = lane >> 4;                 // wave half: 0 or 1
    const int wave = tid >> 5;
    const int icol = tile * TILE_I + wave * 16 + l16;   // this lane's output pixel

    // B operand: i-stripe features, layout B[K = v + 2*hl][N = l16].
    // K order: 0:x 1:y 2:c0 3:c1 4:c2 5..7:zero pad.
    const float xi = (float)(icol % WW) * inv_xy;
    const float yi = (float)(icol / WW) * inv_xy;
    v2f b1, b2;
    if (hl == 0) { b1.x = xi;          b1.y = yi;          b2.x = s_c2[icol]; b2.y = 0.0f; }
    else         { b1.x = s_c0[icol];  b1.y = s_c1[icol];  b2.x = 0.0f;       b2.y = 0.0f; }
    const float sqi = s_sq[icol];

    float acc0 = 0.0f, acc1 = 0.0f, acc2 = 0.0f;

    const float4* sq4 = (const float4*)s_sq;
    const float4* v04 = (const float4*)s_v0;
    const float4* v14 = (const float4*)s_v1;
    const float4* v24 = (const float4*)s_v2;

    for (int jt = 0; jt < PP; jt += 16) {
        // A operand: j-tile features, layout A[M = l16 -> j][K = v + 2*hl].
        const int j  = jt + l16;
        const float xj = (float)(j % WW) * inv_xy;
        const float yj = (float)(j / WW) * inv_xy;
        v2f a1, a2;
        if (hl == 0) { a1.x = xj;        a1.y = yj;        a2.x = s_c2[j]; a2.y = 0.0f; }
        else         { a1.x = s_c0[j];   a1.y = s_c1[j];   a2.x = 0.0f;    a2.y = 0.0f; }

        // Gram tile: g[r] (lane) = f_j[jt + r + 8*hl] . f_i[icol]
        v8f g = {0.f, 0.f, 0.f, 0.f, 0.f, 0.f, 0.f, 0.f};
        g = __builtin_amdgcn_wmma_f32_16x16x4_f32(false, a1, false, b1, (short)0, g, false, false);
        g = __builtin_amdgcn_wmma_f32_16x16x4_f32(false, a2, false, b2, (short)0, g, false, false);

        // Per-lane row block: j = jt + 8*hl + r, r = 0..7 (32B contiguous -> b128 loads)
        const int b4 = (jt + hl * 8) >> 2;
        float4 sjA = sq4[b4], sjB = sq4[b4 + 1];
        float4 w0A = v04[b4], w0B = v04[b4 + 1];
        float4 w1A = v14[b4], w1B = v14[b4 + 1];
        float4 w2A = v24[b4], w2B = v24[b4 + 1];
        float sj[8]  = {sjA.x, sjA.y, sjA.z, sjA.w, sjB.x, sjB.y, sjB.z, sjB.w};
        float vv0[8] = {w0A.x, w0A.y, w0A.z, w0A.w, w0B.x, w0B.y, w0B.z, w0B.w};
        float vv1[8] = {w1A.x, w1A.y, w1A.z, w1A.w, w1B.x, w1B.y, w1B.z, w1B.w};
        float vv2[8] = {w2A.x, w2A.y, w2A.z, w2A.w, w2B.x, w2B.y, w2B.z, w2B.w};

#pragma unroll
        for (int r = 0; r < 8; ++r) {
            float d2 = fmaxf(sqi + sj[r] - 2.0f * g[r], 0.0f);
            float w  = __expf(-0.5f * d2);      // v_exp_f32
            acc0 = fmaf(w, vv0[r], acc0);
            acc1 = fmaf(w, vv1[r], acc1);
            acc2 = fmaf(w, vv2[r], acc2);
        }
    }

    // Combine the two wave halves (each covered half of the j rows).
    acc0 += __shfl_xor(acc0, 16, 32);
    acc1 += __shfl_xor(acc1, 16, 32);
    acc2 += __shfl_xor(acc2, 16, 32);

    // Wave-level max for global normalization.
    float m = fmaxf(fmaxf(acc0, acc1), acc2);
#pragma unroll
    for (int off = 8; off >= 1; off >>= 1) m = fmaxf(m, __shfl_xor(m, off, 32));
    if (lane == 0) atomic_max_float(mx, m);

    if (hl == 0) {
        float* o = out + (size_t)n * KCH * PP;
        o[icol]          = acc0;
        o[PP + icol]     = acc1;
        o[2 * PP + icol] = acc2;
    }
}

__global__ void bf_scale(float* __restrict__ out, const float* __restrict__ mx, int total) {
    int i = blockIdx.x * blockDim.x + threadIdx.x;
    if (i < total) out[i] = out[i] / (*mx);
}

extern "C" void kernel_launch(void* const* d_in, const int* in_sizes, int n_in,
                              void* d_out, int out_size, void* d_ws, size_t ws_size,
                              hipStream_t stream) {
    (void)in_sizes; (void)n_in; (void)out_size; (void)ws_size;
    const float* raw = (const float*)d_in[0];   // images_raw
    const float* nrm = (const float*)d_in[1];   // images_normalized
    float* out = (float*)d_out;
    float* mx  = (float*)d_ws;                  // 1 float of scratch

    const size_t lds_bytes = (size_t)7 * PP * sizeof(float);   // 252 KB (<= 320 KB/WGP)
    // Opt in to >64KB dynamic LDS. Not a stream op -> graph-capture safe; idempotent.
    hipFuncSetAttribute((const void*)bf_main,
                        hipFuncAttributeMaxDynamicSharedMemorySize, (int)lds_bytes);

    bf_init_max<<<1, 1, 0, stream>>>(mx);
    bf_main<<<NIMG * NTILES, THREADS, lds_bytes, stream>>>(raw, nrm, out, mx);

    const int total = NIMG * KCH * PP;
    bf_scale<<<(total + 255) / 256, 256, 0, stream>>>(out, mx, total);
}